// PointConvFlow_70437463654602
// MI455X (gfx1250) — compile-verified
//
#include <hip/hip_runtime.h>
#include <hip/hip_bf16.h>

// PointConvFlow for MI455X (gfx1250, wave32).
//   Kernel A: brute-force KNN (top-16 by squared distance) + inv-dist weights.
//   Kernel B: one wave per query; A-fragments (16x132 feature tile) built
//             directly in registers, weights pre-staged in LDS in WMMA
//             B-fragment layout (one ds_load_b64 per fragment), fp32
//             V_WMMA_F32_16X16X4_F32 pipeline, weighted K-reduce via shfl.

#define B_    8
#define N_    4096
#define D_    64
#define K_    16
#define CIN   131
#define NK1   33    // layer-1 K-steps (132 / 4, zero-padded)
#define NK2   16    // layer-2 K-steps (64 / 4)
#define HSTR  68    // H LDS row stride (even, bank-conflict free)
#define WAVES 2     // waves per block sharing the weight staging

typedef __attribute__((ext_vector_type(2))) float v2f;
typedef __attribute__((ext_vector_type(8))) float v8f;

__device__ __forceinline__ float leaky_(float x) {
  return fmaxf(x, 0.1f * x);  // leaky(x) = max(x, 0.1x)
}

// ---------------------------------------------------------------------------
// Kernel A: KNN. One thread per query point; x2 batch slice cached in LDS.
// Sorted (ascending) top-16 list kept in registers, fully unrolled insert.
// ---------------------------------------------------------------------------
__global__ __launch_bounds__(256) void knn_kernel(
    const float* __restrict__ xyz1, const float* __restrict__ xyz2,
    int* __restrict__ idx_out, float* __restrict__ w_out) {
  __shared__ float s_x2[3 * N_];  // 48 KB

  const int blocks_per_batch = N_ / 256;              // 16
  const int b = blockIdx.x / blocks_per_batch;
  const int chunk = blockIdx.x - b * blocks_per_batch;
  const int tid = threadIdx.x;

  for (int i = tid; i < 3 * N_; i += 256) s_x2[i] = xyz2[b * 3 * N_ + i];
  __syncthreads();

  const int n1 = chunk * 256 + tid;
  const float qx = xyz1[(b * 3 + 0) * N_ + n1];
  const float qy = xyz1[(b * 3 + 1) * N_ + n1];
  const float qz = xyz1[(b * 3 + 2) * N_ + n1];

  float bd[K_];
  int bi[K_];
#pragma unroll
  for (int k = 0; k < K_; ++k) { bd[k] = 3.0e38f; bi[k] = 0; }

  for (int m = 0; m < N_; ++m) {
    const float dx = s_x2[m] - qx;
    const float dy = s_x2[N_ + m] - qy;
    const float dz = s_x2[2 * N_ + m] - qz;
    const float d = fmaf(dx, dx, fmaf(dy, dy, dz * dz));
    if (d < bd[K_ - 1]) {
#pragma unroll
      for (int j = K_ - 1; j >= 1; --j) {
        if (bd[j - 1] > d)      { bd[j] = bd[j - 1]; bi[j] = bi[j - 1]; }
        else if (bd[j] > d)     { bd[j] = d;         bi[j] = m;         }
      }
      if (bd[0] > d)            { bd[0] = d;         bi[0] = m;         }
    }
  }

  float inv[K_];
  float s = 0.0f;
#pragma unroll
  for (int k = 0; k < K_; ++k) {
    const float dist = fmaxf(sqrtf(bd[k]), 1e-10f);
    inv[k] = 1.0f / dist;
    s += inv[k];
  }
  const float rs = 1.0f / s;

  const int q = b * N_ + n1;
#pragma unroll
  for (int k = 0; k < K_; ++k) {
    idx_out[q * K_ + k] = bi[k];
    w_out[q * K_ + k] = inv[k] * rs;
  }
}

// ---------------------------------------------------------------------------
// Kernel B. fp32 WMMA fragment layouts (16x16x4):
//   A (16x4): lanes 0-15 row M=lane hold K=0/1 in v0/v1; lanes 16-31 row
//             M=lane-16 hold K=2/3.
//   B (4x16): v0: lanes 0-15 K=0, lanes 16-31 K=1; v1: K=2 / K=3. Col=lane%16.
//   C/D: VGPR v holds M = v + 8*(lane>=16), N = lane%16.
// Weights are pre-staged in LDS as ready-made B fragments:
//   Wf[(k*4 + nt)*32 + lane] = { WT[4k + h][col], WT[4k + 2 + h][col] }
// with h = lane>>4, col = nt*16 + lane%16 -> one ds_load_b64 per fragment.
// ---------------------------------------------------------------------------
__global__ __launch_bounds__(32 * WAVES) void mlp_kernel(
    const float* __restrict__ xyz1, const float* __restrict__ xyz2,
    const float* __restrict__ p1, const float* __restrict__ p2,
    const float* __restrict__ W1, const float* __restrict__ b1,
    const float* __restrict__ W2, const float* __restrict__ b2,
    const int* __restrict__ knn_idx, const float* __restrict__ knn_w,
    float* __restrict__ out) {
  __shared__ v2f W1f[NK1 * 4 * 32];          // 33792 B, shared by both waves
  __shared__ v2f W2f[NK2 * 4 * 32];          // 16384 B, shared by both waves
  __shared__ float Hs[WAVES][16 * HSTR];     // 2 x 4352 B, per-wave H tile

  const int tid = threadIdx.x;
  const int lane = tid & 31;
  const int wv = tid >> 5;
  const int q = blockIdx.x * WAVES + wv;     // query id
  const int b = q >> 12;
  const int n1 = q & (N_ - 1);

  // ---- stage W1 / W2 into LDS in B-fragment layout (block-cooperative) ----
  for (int e = tid; e < NK1 * 4 * 32; e += 32 * WAVES) {
    const int l = e & 31, nt = (e >> 5) & 3, k = e >> 7;
    const int col = nt * 16 + (l & 15), h = l >> 4;
    const int c0 = 4 * k + h, c1 = 4 * k + 2 + h;
    v2f w;
    w.x = (c0 < CIN) ? W1[col * CIN + c0] : 0.0f;
    w.y = (c1 < CIN) ? W1[col * CIN + c1] : 0.0f;
    W1f[e] = w;
  }
  for (int e = tid; e < NK2 * 4 * 32; e += 32 * WAVES) {
    const int l = e & 31, nt = (e >> 5) & 3, k = e >> 7;
    const int col = nt * 16 + (l & 15), h = l >> 4;
    v2f w;
    w.x = W2[col * 64 + 4 * k + h];
    w.y = W2[col * 64 + 4 * k + 2 + h];
    W2f[e] = w;
  }
  __syncthreads();

  const int mrow = lane & 15;
  const int hi = lane >> 4;                  // 0: lanes 0-15, 1: lanes 16-31

  // ---- build layer-1 A fragments in registers ----
  // Lane owns row M=mrow and the column-parity class {4k+2*hi, 4k+2*hi+1}.
  const int im = knn_idx[q * K_ + mrow];     // neighbor index of this row
  v2f af[NK1];
#pragma unroll
  for (int k = 0; k < 16; ++k) {             // cols 0..63: p1 (broadcast)
    const int c0 = 4 * k + 2 * hi;
    af[k].x = p1[((b * 64 + c0) << 12) + n1];
    af[k].y = p1[((b * 64 + c0 + 1) << 12) + n1];
  }
#pragma unroll
  for (int k = 16; k < 32; ++k) {            // cols 64..127: gathered p2
    const int c0 = 4 * k + 2 * hi - 64;
    af[k].x = p2[((b * 64 + c0) << 12) + im];
    af[k].y = p2[((b * 64 + c0 + 1) << 12) + im];
  }
  {                                          // cols 128..130: dir_xyz, 131: 0
    const int c0 = hi ? 2 : 0;
    af[32].x = xyz2[((b * 3 + c0) << 12) + im] - xyz1[((b * 3 + c0) << 12) + n1];
    af[32].y = hi ? 0.0f
                  : (xyz2[((b * 3 + 1) << 12) + im] -
                     xyz1[((b * 3 + 1) << 12) + n1]);
  }

  // ---- layer 1: 33 K-steps x 4 N-tiles ----
  v8f acc[4] = {};
#pragma unroll
  for (int k = 0; k < NK1; ++k) {
#pragma unroll
    for (int nt = 0; nt < 4; ++nt) {
      const v2f bb = W1f[(k * 4 + nt) * 32 + lane];
      acc[nt] = __builtin_amdgcn_wmma_f32_16x16x4_f32(
          false, af[k], false, bb, (short)0, acc[nt], false, false);
    }
  }

  // ---- bias + leaky, transpose H into per-wave LDS tile ----
  float* hp = &Hs[wv][0];
#pragma unroll
  for (int nt = 0; nt < 4; ++nt) {
    const int col = nt * 16 + mrow;
    const float bv = b1[col];
#pragma unroll
    for (int v = 0; v < 8; ++v) {
      hp[(v + 8 * hi) * HSTR + col] = leaky_(acc[nt][v] + bv);
    }
  }
  __syncthreads();

  // ---- layer 2: 16 K-steps x 4 N-tiles ----
  v8f acc2[4] = {};
#pragma unroll
  for (int k = 0; k < NK2; ++k) {
    const v2f a = *(const v2f*)&hp[mrow * HSTR + 4 * k + 2 * hi];
#pragma unroll
    for (int nt = 0; nt < 4; ++nt) {
      const v2f bb = W2f[(k * 4 + nt) * 32 + lane];
      acc2[nt] = __builtin_amdgcn_wmma_f32_16x16x4_f32(
          false, a, false, bb, (short)0, acc2[nt], false, false);
    }
  }

  // ---- bias + leaky + inverse-distance weighted reduce over the 16 rows ----
  float wreg[8];
#pragma unroll
  for (int v = 0; v < 8; ++v) wreg[v] = knn_w[q * K_ + v + 8 * hi];

#pragma unroll
  for (int nt = 0; nt < 4; ++nt) {
    const int col = nt * 16 + mrow;
    const float bv = b2[col];
    float p = 0.0f;
#pragma unroll
    for (int v = 0; v < 8; ++v) {
      p += wreg[v] * leaky_(acc2[nt][v] + bv);
    }
    p += __shfl_xor(p, 16);                  // merge the two M-halves
    if (hi == 0) out[((b * 64 + col) << 12) + n1] = p;
  }
}

// ---------------------------------------------------------------------------
extern "C" void kernel_launch(void* const* d_in, const int* in_sizes, int n_in,
                              void* d_out, int out_size, void* d_ws, size_t ws_size,
                              hipStream_t stream) {
  (void)in_sizes; (void)n_in; (void)out_size; (void)ws_size;
  const float* xyz1 = (const float*)d_in[0];
  const float* xyz2 = (const float*)d_in[1];
  const float* points1 = (const float*)d_in[2];
  const float* points2 = (const float*)d_in[3];
  const float* W1 = (const float*)d_in[4];
  const float* b1 = (const float*)d_in[5];
  const float* W2 = (const float*)d_in[6];
  const float* b2 = (const float*)d_in[7];
  float* out = (float*)d_out;

  int* knn_idx = (int*)d_ws;
  float* knn_w = (float*)((char*)d_ws + (size_t)B_ * N_ * K_ * sizeof(int));

  knn_kernel<<<dim3(B_ * (N_ / 256)), dim3(256), 0, stream>>>(
      xyz1, xyz2, knn_idx, knn_w);

  mlp_kernel<<<dim3(B_ * N_ / WAVES), dim3(32 * WAVES), 0, stream>>>(
      xyz1, xyz2, points1, points2, W1, b1, W2, b2, knn_idx, knn_w, out);
}